// CLIP_text_image_concat_46755013984821
// MI455X (gfx1250) — compile-verified
//
#include <hip/hip_runtime.h>
#include <hip/hip_bf16.h>
#include <cstdint>

// CDNA5 / gfx1250 fused CLIP per-class heads.
// Main GEMM (img[256x768] @ W1img[k][768x312] per head) runs on
// V_WMMA_F32_16X16X4_F32. Operand staging uses double-buffered
// GLOBAL_LOAD_ASYNC_TO_LDS_B128 (ASYNCcnt) so the next stage's fill
// overlaps the current stage's WMMAs; W1 (817 MB single-use stream) is
// loaded non-temporal so the shared img tile stays resident in L2.
// BatchNorm (train-mode batch stats) + W2 dot are fused into the GEMM
// epilogue since each workgroup holds the full batch dimension.

typedef __attribute__((ext_vector_type(2))) float v2f;
typedef __attribute__((ext_vector_type(8))) float v8f;

#define BATCH 256
#define IMG_D 768
#define CAT_D 1280
#define TXT_D 512
#define HIDN  312
#define NBLK  64     // HID columns per workgroup
#define NNB   5      // ceil(312/64)
#define KB    16     // K-depth staged per iteration
#define NSTG  (IMG_D / KB)
#define LDA   20     // LDS A row stride (floats): conflict-free, 16B-aligned
#define LDB   72     // LDS B row stride (floats): halves hit disjoint banks
#define BN_EPS 1e-5f

// ---------------------------------------------------------------------------
// prep: per head, tvec[k][h] = sum_d txt[k,d]*W1[k,768+d,h] + b1[k,h]
//       txtdot[k] = sum_d txt[k,d]*linW[k,768+d]
// ---------------------------------------------------------------------------
__global__ void prep_kernel(const float* __restrict__ txt,   // [K][512]
                            const float* __restrict__ W1,    // [K][1280][312]
                            const float* __restrict__ b1,    // [K][312]
                            const float* __restrict__ linW,  // [K][1280]
                            float* __restrict__ tvec,        // [K][312]
                            float* __restrict__ txtdot)      // [K]
{
    __shared__ float sred[320];
    const int k = blockIdx.x;
    const int t = threadIdx.x;                 // 320 threads
    const float* txtk = txt + (size_t)k * TXT_D;
    if (t < HIDN) {
        const float* W1t = W1 + ((size_t)k * CAT_D + IMG_D) * HIDN + t;
        float a = 0.f;
        #pragma unroll 4
        for (int d = 0; d < TXT_D; ++d)
            a = fmaf(txtk[d], W1t[(size_t)d * HIDN], a);
        tvec[(size_t)k * HIDN + t] = a + b1[(size_t)k * HIDN + t];
    }
    // linear-head text dot (block reduction)
    const float* lwt = linW + (size_t)k * CAT_D + IMG_D;
    float p = 0.f;
    for (int d = t; d < TXT_D; d += 320) p = fmaf(txtk[d], lwt[d], p);
    sred[t] = p;
    __syncthreads();
    if (t == 0) {
        float s = 0.f;
        for (int i = 0; i < 320; ++i) s += sred[i];
        txtdot[k] = s;
    }
}

// ---------------------------------------------------------------------------
// lin: lin[b][k] = img[b,:768] . linW[k,:768] + txtdot[k] + lin_b[k]
// thread = batch row, 8 heads per block; linW loads are wave-uniform.
// K is a multiple of 8 for both vocabularies (312, 200).
// ---------------------------------------------------------------------------
__global__ void lin_kernel(const float* __restrict__ img,
                           const float* __restrict__ linW,   // [K][1280]
                           const float* __restrict__ lin_b,
                           const float* __restrict__ txtdot,
                           float* __restrict__ lin_out,      // [256][K]
                           int K)
{
    const int b  = threadIdx.x;                // 256
    const int k0 = blockIdx.x * 8;
    float acc[8] = {0.f,0.f,0.f,0.f,0.f,0.f,0.f,0.f};
    const float4* imgr = (const float4*)(img + (size_t)b * IMG_D);
    for (int d4 = 0; d4 < IMG_D / 4; ++d4) {
        float4 iv = imgr[d4];
        #pragma unroll
        for (int kk = 0; kk < 8; ++kk) {
            float4 wv = *(const float4*)(linW + (size_t)(k0 + kk) * CAT_D + d4 * 4);
            acc[kk] = fmaf(iv.x, wv.x, fmaf(iv.y, wv.y,
                      fmaf(iv.z, wv.z, fmaf(iv.w, wv.w, acc[kk]))));
        }
    }
    #pragma unroll
    for (int kk = 0; kk < 8; ++kk) {
        int k = k0 + kk;
        lin_out[(size_t)b * K + k] = acc[kk] + txtdot[k] + lin_b[k];
    }
}

// ---------------------------------------------------------------------------
// fused GEMM + ReLU + BatchNorm(batch stats) + W2 dot.
// grid = (nblock 0..4, head k); block = 256 threads = 8 waves.
// Wave w owns batch rows [32w, 32w+32) as two 16-row WMMA tiles x 4 N-tiles.
// Double-buffered async-to-LDS staging overlaps fills with WMMA compute.
// Writes per-nblock partial classifier pcls[nb][b][k] (deterministic; a
// final kernel sums the 5 partials).
// ---------------------------------------------------------------------------
__global__ void fused_head_kernel(const float* __restrict__ img,   // [256][768]
                                  const float* __restrict__ W1,    // [K][1280][312]
                                  const float* __restrict__ tvec,  // [K][312]
                                  const float* __restrict__ gamma_,
                                  const float* __restrict__ beta_,
                                  const float* __restrict__ W2,    // [K][312]
                                  float* __restrict__ pcls,        // [5][256][K]
                                  int K)
{
    __shared__ float lA[2 * BATCH * LDA];   // 2 x 20 KB
    __shared__ float lB[2 * KB * LDB];      // 2 x 4.5 KB
    __shared__ float red1[8 * NBLK];
    __shared__ float red2[8 * NBLK];
    __shared__ float cw_s[NBLK];
    __shared__ float shiftdot_s;

    const int k     = blockIdx.y;
    const int n0    = blockIdx.x * NBLK;
    const int tid   = threadIdx.x;
    const int wave  = tid >> 5;
    const int lane  = tid & 31;
    const int lhalf = lane >> 4;       // 0 | 1
    const int lmod  = lane & 15;
    const int m0    = wave * 32;

    const float* W1k = W1 + (size_t)k * CAT_D * HIDN;
    const int kdB = tid >> 4;            // B-stage row this thread fills
    const int nnB = (tid & 15) * 4;      // B-stage column quad

    v8f acc[2][4];
    #pragma unroll
    for (int mt = 0; mt < 2; ++mt)
        #pragma unroll
        for (int nt = 0; nt < 4; ++nt)
            acc[mt][nt] = (v8f)0.f;

    // ---- async stage issue: A row (4 x b128, same offsets both sides) + B quad
    auto issue_stage = [&](int buf, int kk) {
        uint32_t la = (uint32_t)(uintptr_t)(lA + buf * (BATCH * LDA) + tid * LDA);
        const float* ga = img + (size_t)tid * IMG_D + kk;
        asm volatile(
            "global_load_async_to_lds_b128 %0, %1, off offset:0\n\t"
            "global_load_async_to_lds_b128 %0, %1, off offset:16\n\t"
            "global_load_async_to_lds_b128 %0, %1, off offset:32\n\t"
            "global_load_async_to_lds_b128 %0, %1, off offset:48"
            :: "v"(la), "v"(ga) : "memory");
        uint32_t lb = (uint32_t)(uintptr_t)(lB + buf * (KB * LDB) + kdB * LDB + nnB);
        const float* gb = W1k + (size_t)(kk + kdB) * HIDN + n0 + nnB;
        asm volatile(
            "global_load_async_to_lds_b128 %0, %1, off offset:0 th:TH_LOAD_NT"
            :: "v"(lb), "v"(gb) : "memory");
    };

    issue_stage(0, 0);
    for (int it = 0; it < NSTG; ++it) {
        const int cur = it & 1;
        asm volatile("s_wait_asynccnt 0" ::: "memory");
        __syncthreads();                         // all waves' fills of `cur` done;
                                                 // all waves done reading `cur^1`
        if (it + 1 < NSTG) issue_stage(cur ^ 1, (it + 1) * KB);

        const float* lAc = lA + cur * (BATCH * LDA);
        const float* lBc = lB + cur * (KB * LDB);
        #pragma unroll
        for (int ks = 0; ks < KB; ks += 4) {
            v2f a0 = *(const v2f*)(lAc + (m0 + lmod) * LDA + ks + 2 * lhalf);
            v2f a1 = *(const v2f*)(lAc + (m0 + 16 + lmod) * LDA + ks + 2 * lhalf);
            #pragma unroll
            for (int nt = 0; nt < 4; ++nt) {
                v2f b;
                b.x = lBc[(ks + 2 * lhalf)     * LDB + nt * 16 + lmod];
                b.y = lBc[(ks + 2 * lhalf + 1) * LDB + nt * 16 + lmod];
                acc[0][nt] = __builtin_amdgcn_wmma_f32_16x16x4_f32(
                    false, a0, false, b, (short)0, acc[0][nt], false, false);
                acc[1][nt] = __builtin_amdgcn_wmma_f32_16x16x4_f32(
                    false, a1, false, b, (short)0, acc[1][nt], false, false);
            }
        }
    }

    // ---- epilogue: bias(text proj) + ReLU + batch stats ----
    float tv[4];
    #pragma unroll
    for (int nt = 0; nt < 4; ++nt) {
        int n = n0 + nt * 16 + lmod;
        tv[nt] = (n < HIDN) ? tvec[(size_t)k * HIDN + n] : 0.f;
    }
    float s1[4] = {0.f,0.f,0.f,0.f};
    float s2[4] = {0.f,0.f,0.f,0.f};
    #pragma unroll
    for (int mt = 0; mt < 2; ++mt)
        #pragma unroll
        for (int nt = 0; nt < 4; ++nt)
            #pragma unroll
            for (int j = 0; j < 8; ++j) {
                float x = fmaxf(acc[mt][nt][j] + tv[nt], 0.f);
                acc[mt][nt][j] = x;
                s1[nt] += x;
                s2[nt] = fmaf(x, x, s2[nt]);
            }
    #pragma unroll
    for (int nt = 0; nt < 4; ++nt) {
        s1[nt] += __shfl_xor(s1[nt], 16, 32);
        s2[nt] += __shfl_xor(s2[nt], 16, 32);
    }
    __syncthreads();          // main-loop LDS buffers are dead; safe to reuse phase
    if (lane < 16) {
        #pragma unroll
        for (int nt = 0; nt < 4; ++nt) {
            red1[wave * NBLK + nt * 16 + lmod] = s1[nt];
            red2[wave * NBLK + nt * 16 + lmod] = s2[nt];
        }
    }
    __syncthreads();

    if (tid < NBLK) {
        float S1 = 0.f, S2 = 0.f;
        #pragma unroll
        for (int w = 0; w < 8; ++w) { S1 += red1[w * NBLK + tid]; S2 += red2[w * NBLK + tid]; }
        float mean = S1 * (1.f / BATCH);
        float var  = S2 * (1.f / BATCH) - mean * mean;
        int n = n0 + tid;
        float g = 0.f, bb = 0.f, w2 = 0.f;
        if (n < HIDN) {
            g  = gamma_[(size_t)k * HIDN + n];
            bb = beta_ [(size_t)k * HIDN + n];
            w2 = W2    [(size_t)k * HIDN + n];
        }
        float scale = g * rsqrtf(var + BN_EPS);
        cw_s[tid] = scale * w2;
        red1[tid] = (bb - mean * scale) * w2;   // shift * w2 (0 if padded col)
    }
    __syncthreads();
    if (tid == 0) {
        float s = 0.f;
        for (int i = 0; i < NBLK; ++i) s += red1[i];
        shiftdot_s = s;
    }
    __syncthreads();

    // ---- per-row dot with folded BN*W2 coefficients ----
    float cwv[4];
    #pragma unroll
    for (int nt = 0; nt < 4; ++nt) cwv[nt] = cw_s[nt * 16 + lmod];

    const float sdot = shiftdot_s;
    #pragma unroll
    for (int mt = 0; mt < 2; ++mt) {
        float rd[8];
        #pragma unroll
        for (int j = 0; j < 8; ++j)
            rd[j] = acc[mt][0][j] * cwv[0] + acc[mt][1][j] * cwv[1]
                  + acc[mt][2][j] * cwv[2] + acc[mt][3][j] * cwv[3];
        #pragma unroll
        for (int off = 1; off < 16; off <<= 1)
            #pragma unroll
            for (int j = 0; j < 8; ++j)
                rd[j] += __shfl_xor(rd[j], off, 32);
        if (lmod == 0) {
            int rb = m0 + mt * 16 + 8 * lhalf;
            #pragma unroll
            for (int j = 0; j < 8; ++j)
                pcls[(size_t)(blockIdx.x * BATCH + rb + j) * K + k] = rd[j] + sdot;
        }
    }
}

// ---------------------------------------------------------------------------
// reduce: cls[b][k] = b2[k] + sum_nb pcls[nb][b][k]   (deterministic)
// ---------------------------------------------------------------------------
__global__ void reduce_cls_kernel(const float* __restrict__ pcls,
                                  const float* __restrict__ b2,
                                  float* __restrict__ cls,
                                  int K)
{
    int idx = blockIdx.x * 256 + threadIdx.x;      // b*K + k
    if (idx < BATCH * K) {
        int k = idx % K;
        float s = b2[k];
        #pragma unroll
        for (int nb = 0; nb < NNB; ++nb)
            s += pcls[(size_t)nb * BATCH * K + idx];
        cls[idx] = s;
    }
}

extern "C" void kernel_launch(void* const* d_in, const int* in_sizes, int n_in,
                              void* d_out, int out_size, void* d_ws, size_t ws_size,
                              hipStream_t stream) {
    (void)in_sizes; (void)n_in; (void)out_size; (void)ws_size;
    const float* img        = (const float*)d_in[0];
    const float* adj_txt    = (const float*)d_in[1];
    const float* noun_txt   = (const float*)d_in[2];
    const float* adj_linW   = (const float*)d_in[3];
    const float* adj_linb   = (const float*)d_in[4];
    const float* adj_W1     = (const float*)d_in[5];
    const float* adj_b1     = (const float*)d_in[6];
    const float* adj_gamma  = (const float*)d_in[7];
    const float* adj_beta   = (const float*)d_in[8];
    const float* adj_W2     = (const float*)d_in[9];
    const float* adj_b2     = (const float*)d_in[10];
    const float* noun_linW  = (const float*)d_in[11];
    const float* noun_linb  = (const float*)d_in[12];
    const float* noun_W1    = (const float*)d_in[13];
    const float* noun_b1    = (const float*)d_in[14];
    const float* noun_gamma = (const float*)d_in[15];
    const float* noun_beta  = (const float*)d_in[16];
    const float* noun_W2    = (const float*)d_in[17];
    const float* noun_b2    = (const float*)d_in[18];

    const int KA = 312, KN = 200;

    float* out        = (float*)d_out;
    float* adj_lin_o  = out;                               // 256*312
    float* noun_lin_o = adj_lin_o  + BATCH * KA;           // 256*200
    float* adj_cls_o  = noun_lin_o + BATCH * KN;           // 256*312
    float* noun_cls_o = adj_cls_o  + BATCH * KA;           // 256*200

    float* ws        = (float*)d_ws;
    float* adj_tvec  = ws;                                 // 312*312
    float* noun_tvec = adj_tvec  + KA * HIDN;              // 200*312
    float* adj_td    = noun_tvec + KN * HIDN;              // 312
    float* noun_td   = adj_td    + KA;                     // 200
    float* adj_pcls  = noun_td   + KN;                     // 5*256*312
    float* noun_pcls = adj_pcls  + NNB * BATCH * KA;       // 5*256*200

    // phase 1: text projections + linear-head text dots
    prep_kernel<<<KA, 320, 0, stream>>>(adj_txt,  adj_W1,  adj_b1,  adj_linW,  adj_tvec,  adj_td);
    prep_kernel<<<KN, 320, 0, stream>>>(noun_txt, noun_W1, noun_b1, noun_linW, noun_tvec, noun_td);

    // phase 2: tiny linear heads
    lin_kernel<<<KA / 8, BATCH, 0, stream>>>(img, adj_linW,  adj_linb,  adj_td,  adj_lin_o,  KA);
    lin_kernel<<<KN / 8, BATCH, 0, stream>>>(img, noun_linW, noun_linb, noun_td, noun_lin_o, KN);

    // phase 3: fused WMMA GEMM + ReLU + BatchNorm + W2 dot (per-nblock partials)
    fused_head_kernel<<<dim3(NNB, KA), BATCH, 0, stream>>>(
        img, adj_W1,  adj_tvec,  adj_gamma,  adj_beta,  adj_W2,  adj_pcls,  KA);
    fused_head_kernel<<<dim3(NNB, KN), BATCH, 0, stream>>>(
        img, noun_W1, noun_tvec, noun_gamma, noun_beta, noun_W2, noun_pcls, KN);

    // phase 4: deterministic partial-sum reduction + b2
    reduce_cls_kernel<<<(BATCH * KA + 255) / 256, 256, 0, stream>>>(adj_pcls,  adj_b2,  adj_cls_o,  KA);
    reduce_cls_kernel<<<(BATCH * KN + 255) / 256, 256, 0, stream>>>(noun_pcls, noun_b2, noun_cls_o, KN);
}